// LocalAttn_14697378087719
// MI455X (gfx1250) — compile-verified
//
#include <hip/hip_runtime.h>
#include <hip/hip_bf16.h>

typedef __attribute__((ext_vector_type(2))) float v2f;
typedef __attribute__((ext_vector_type(8))) float v8f;

#define EPSV 1e-5f
constexpr int Bn = 2, Cc = 256, Hh = 128, Ww = 128, Gg = 8;
constexpr int HW = Hh * Ww;            // 16384 pixels per image

// ---------------------------------------------------------------------------
// Kernel A: t = tanh(BN(gconv1x1(x, w1) + b1))
// Per group g: T[8 x P] = W1[g](8x32) @ X[g](32 x P), via WMMA f32 16x16x4.
// Block = 256 thr = 8 waves; wave w handles group w for one 16-pixel tile.
// All fragments loaded up front (single load clause), then 8 chained WMMAs.
// ---------------------------------------------------------------------------
__global__ __launch_bounds__(256)
void conv1_bn_tanh(const float* __restrict__ x, const float* __restrict__ w1,
                   const float* __restrict__ b1, const float* __restrict__ g1,
                   const float* __restrict__ be1, const float* __restrict__ m1,
                   const float* __restrict__ v1, float* __restrict__ t)
{
    const int lane  = threadIdx.x & 31;
    const int g     = threadIdx.x >> 5;          // wave id -> group
    const int tile  = blockIdx.x;                // 0 .. Bn*1024-1
    const int b     = tile >> 10;
    const int pbase = (tile & 1023) * 16;

    const int mrow  = lane & 15;                 // M (A) or N (B) index
    const int khalf = (lane >> 4) * 2;           // K sub-offset per half-wave
    const int arow  = mrow & 7;                  // clamped weight row (branch-free)
    const bool aval = (mrow < 8);

    const float* xg = x  + ((size_t)b * Cc + g * 32) * HW;
    const float* wg = w1 + g * 8 * 32;           // [o][i]

    v2f af[8], bf[8];
#pragma unroll
    for (int kk = 0; kk < 8; ++kk) {             // K = 32 in steps of 4
        const int k0 = kk * 4 + khalf;
        // A 16x4: rows = out channel (0..7 valid; rows 8..15 forced to zero)
        const float ax = wg[arow * 32 + k0];
        const float ay = wg[arow * 32 + k0 + 1];
        af[kk].x = aval ? ax : 0.f;
        af[kk].y = aval ? ay : 0.f;
        // B 4x16: rows = in channel, cols = pixel
        bf[kk].x = xg[(size_t)k0       * HW + pbase + mrow];
        bf[kk].y = xg[(size_t)(k0 + 1) * HW + pbase + mrow];
    }
    v8f acc = {};
#pragma unroll
    for (int kk = 0; kk < 8; ++kk)
        acc = __builtin_amdgcn_wmma_f32_16x16x4_f32(false, af[kk], false, bf[kk],
                                                    (short)0, acc, false, false);

    // C/D: VGPR i, lanes 0-15 -> M=i (valid rows 0..7), lanes 16-31 -> M=i+8 (unused)
    if (lane < 16) {
        const int pix = pbase + lane;
#pragma unroll
        for (int i = 0; i < 8; ++i) {
            const int ch = g * 8 + i;
            const float inv = g1[ch] * rsqrtf(v1[ch] + EPSV);
            const float val = (acc[i] + b1[ch]) * inv + (be1[ch] - m1[ch] * inv);
            t[((size_t)b * 64 + ch) * HW + pix] = tanhf(val);
        }
    }
}

// ---------------------------------------------------------------------------
// Kernel B: mn = BN(gconv1x1(t, w2) + b2)   (8 -> 10 per group; tiny: scalar)
// ---------------------------------------------------------------------------
__global__ __launch_bounds__(256)
void conv2_bn(const float* __restrict__ t, const float* __restrict__ w2,
              const float* __restrict__ b2, const float* __restrict__ g2,
              const float* __restrict__ be2, const float* __restrict__ m2,
              const float* __restrict__ v2, float* __restrict__ mn)
{
    __shared__ float w2s[8 * 10 * 8];
    for (int i = threadIdx.x; i < 640; i += 256) w2s[i] = w2[i];
    __syncthreads();

    const int tid = blockIdx.x * 256 + threadIdx.x;   // 0 .. Bn*HW-1
    const int b = tid / HW;
    const int p = tid % HW;
    const float* tb = t  + (size_t)b * 64 * HW + p;
    float*       mb = mn + (size_t)b * 80 * HW + p;

#pragma unroll
    for (int g = 0; g < 8; ++g) {
        float tin[8];
#pragma unroll
        for (int i = 0; i < 8; ++i) tin[i] = tb[(size_t)(g * 8 + i) * HW];
#pragma unroll
        for (int o = 0; o < 10; ++o) {
            float acc = 0.f;
#pragma unroll
            for (int i = 0; i < 8; ++i) acc += tin[i] * w2s[(g * 10 + o) * 8 + i];
            const int ch = g * 10 + o;
            const float inv = g2[ch] * rsqrtf(v2[ch] + EPSV);
            mb[(size_t)ch * HW] = (acc + b2[ch]) * inv + (be2[ch] - m2[ch] * inv);
        }
    }
}

// ---------------------------------------------------------------------------
// Kernel C: v = gconv1x1(x, wv)  (32 -> 32 per group; main WMMA GEMM)
// Per group: V[32 x P] = Wv[g](32x32) @ X[g](32 x P); M=32 via two accumulators.
// Block = 8 waves; wave = one 16-pixel tile; block = (b, g, strip of 8 tiles).
// Fragments loaded up front, then 16 back-to-back WMMAs.
// ---------------------------------------------------------------------------
__global__ __launch_bounds__(256)
void vconv(const float* __restrict__ x, const float* __restrict__ wv,
           float* __restrict__ v)
{
    const int lane = threadIdx.x & 31;
    const int wid  = threadIdx.x >> 5;
    int blk = blockIdx.x;                        // Bn*Gg*128 blocks
    const int strip = blk & 127;  blk >>= 7;
    const int g     = blk & 7;    blk >>= 3;
    const int b     = blk;
    const int pbase = (strip * 8 + wid) * 16;

    const int mrow  = lane & 15;
    const int khalf = (lane >> 4) * 2;

    const float* xg = x  + ((size_t)b * Cc + g * 32) * HW;
    const float* wg = wv + g * 32 * 32;          // [o][i]

    v2f alo[8], ahi[8], bf[8];
#pragma unroll
    for (int kk = 0; kk < 8; ++kk) {
        const int k0 = kk * 4 + khalf;
        bf[kk].x  = xg[(size_t)k0       * HW + pbase + mrow];
        bf[kk].y  = xg[(size_t)(k0 + 1) * HW + pbase + mrow];
        alo[kk].x = wg[mrow * 32 + k0];         alo[kk].y = wg[mrow * 32 + k0 + 1];
        ahi[kk].x = wg[(mrow + 16) * 32 + k0];  ahi[kk].y = wg[(mrow + 16) * 32 + k0 + 1];
    }
    v8f clo = {}, chi = {};
#pragma unroll
    for (int kk = 0; kk < 8; ++kk) {
        clo = __builtin_amdgcn_wmma_f32_16x16x4_f32(false, alo[kk], false, bf[kk],
                                                    (short)0, clo, false, false);
        chi = __builtin_amdgcn_wmma_f32_16x16x4_f32(false, ahi[kk], false, bf[kk],
                                                    (short)0, chi, false, false);
    }
    const int orow = (lane < 16) ? 0 : 8;
    const int pix  = pbase + mrow;
    float* vg = v + ((size_t)b * Cc + g * 32) * HW + pix;
#pragma unroll
    for (int i = 0; i < 8; ++i) {
        vg[(size_t)(i + orow)      * HW] = clo[i];   // out ch 0..15
        vg[(size_t)(i + orow + 16) * HW] = chi[i];   // out ch 16..31
    }
}

// ---------------------------------------------------------------------------
// Kernel D: 3x3 local attention. Block = 16x16 pixel tile for one (b, g).
// Stage neighbor(18x18) and v-tile(32x18x18) in LDS (zero-padded), then
// per-pixel softmax over 9 taps and weighted aggregation of 32 channels.
// ---------------------------------------------------------------------------
__global__ __launch_bounds__(256)
void attn_kernel(const float* __restrict__ mn, const float* __restrict__ v,
                 float* __restrict__ out)
{
    __shared__ float vs[32][18 * 18];   // 41472 B
    __shared__ float nbs[18 * 18];      //  1296 B

    const int bg = blockIdx.x;
    const int b  = bg >> 3, g = bg & 7;
    const int y0 = blockIdx.y * 16, x0 = blockIdx.z * 16;
    const int lx = threadIdx.x & 15, ly = threadIdx.x >> 4;

    const float* vgrp = v  + ((size_t)b * 256 + g * 32) * HW;
    const float* nbp  = mn + ((size_t)b * 80 + g) * HW;

    for (int e = threadIdx.x; e < 324; e += 256) {
        const int r = e / 18, cdx = e % 18;
        const int gy = y0 - 1 + r, gx = x0 - 1 + cdx;
        nbs[e] = (gy >= 0 && gy < Hh && gx >= 0 && gx < Ww) ? nbp[gy * Ww + gx] : 0.f;
    }
    for (int e = threadIdx.x; e < 32 * 324; e += 256) {
        const int c = e / 324, rem = e % 324;
        const int r = rem / 18, cdx = rem % 18;
        const int gy = y0 - 1 + r, gx = x0 - 1 + cdx;
        vs[c][rem] = (gy >= 0 && gy < Hh && gx >= 0 && gx < Ww)
                         ? vgrp[(size_t)c * HW + gy * Ww + gx] : 0.f;
    }
    __syncthreads();

    const int y = y0 + ly, xp = x0 + lx;
    const float* maskp = mn + ((size_t)b * 80 + 8 + g * 9) * HW + (size_t)y * Ww + xp;

    float lg[9];
#pragma unroll
    for (int k = 0; k < 9; ++k) {
        const int i = k / 3, j = k % 3;
        lg[k] = maskp[(size_t)k * HW] + nbs[(ly + i) * 18 + (lx + j)];
    }
    float mx = lg[0];
#pragma unroll
    for (int k = 1; k < 9; ++k) mx = fmaxf(mx, lg[k]);
    float s = 0.f;
#pragma unroll
    for (int k = 0; k < 9; ++k) { lg[k] = expf(lg[k] - mx); s += lg[k]; }
    const float sinv = 1.f / s;
#pragma unroll
    for (int k = 0; k < 9; ++k) lg[k] *= sinv;

    float* outp = out + ((size_t)b * 256 + g * 32) * HW + (size_t)y * Ww + xp;
#pragma unroll
    for (int c = 0; c < 32; ++c) {
        float acc = 0.f;
#pragma unroll
        for (int k = 0; k < 9; ++k) {
            const int i = k / 3, j = k % 3;
            acc += lg[k] * vs[c][(ly + i) * 18 + (lx + j)];
        }
        outp[(size_t)c * HW] = acc;
    }
}

// ---------------------------------------------------------------------------
extern "C" void kernel_launch(void* const* d_in, const int* in_sizes, int n_in,
                              void* d_out, int out_size, void* d_ws, size_t ws_size,
                              hipStream_t stream)
{
    const float* x   = (const float*)d_in[0];
    const float* w1  = (const float*)d_in[1];
    const float* b1  = (const float*)d_in[2];
    const float* g1  = (const float*)d_in[3];
    const float* be1 = (const float*)d_in[4];
    const float* m1  = (const float*)d_in[5];
    const float* v1  = (const float*)d_in[6];
    const float* w2  = (const float*)d_in[7];
    const float* b2  = (const float*)d_in[8];
    const float* g2  = (const float*)d_in[9];
    const float* be2 = (const float*)d_in[10];
    const float* m2  = (const float*)d_in[11];
    const float* v2  = (const float*)d_in[12];
    const float* wvp = (const float*)d_in[13];
    float* out = (float*)d_out;

    float* t    = (float*)d_ws;                        // Bn*64*HW  ( 8.4 MB)
    float* mnb  = t   + (size_t)Bn * 64 * HW;          // Bn*80*HW  (10.5 MB)
    float* vbuf = mnb + (size_t)Bn * 80 * HW;          // Bn*256*HW (33.5 MB)

    conv1_bn_tanh<<<Bn * 1024, 256, 0, stream>>>(x, w1, b1, g1, be1, m1, v1, t);
    conv2_bn<<<(Bn * HW) / 256, 256, 0, stream>>>(t, w2, b2, g2, be2, m2, v2, mnb);
    vconv<<<Bn * Gg * 128, 256, 0, stream>>>(x, wvp, vbuf);
    attn_kernel<<<dim3(Bn * Gg, Hh / 16, Ww / 16), 256, 0, stream>>>(mnb, vbuf, out);
}